// NODE_64544768525258
// MI455X (gfx1250) — compile-verified
//
#include <hip/hip_runtime.h>
#include <math.h>

typedef float v2f __attribute__((ext_vector_type(2)));
typedef float v8f __attribute__((ext_vector_type(8)));

#define BB 512
#define MM 256
#define INDIM 50

// workspace float offsets
#define O_D    0u         // [4*512][256] raw d_i[b,m] (scaleT folded)
#define O_PHI  524288u    // [4*512][256] phi
#define O_INP  1048576u   // [4*512][64]  inp_i[b,k] (k>=50 zero)
#define O_WV   1179648u   // [4*512] c_i*grad_x/B
#define O_WV2  1181696u   // [4*512] c_i*grad_x*scaleT/B
#define O_GX   1183744u   // [512]
#define O_GB   1184256u   // [512]

__global__ __launch_bounds__(128) void k1_stages(
    const float* __restrict__ U0, const float* __restrict__ U1, const float* __restrict__ U2,
    const float* __restrict__ S,  const float* __restrict__ w,
    const float* __restrict__ b_t_n, const float* __restrict__ b_obs,
    const int* __restrict__ b_i_n,
    float* __restrict__ ws, float* __restrict__ out)
{
    __shared__ float S_s[256 * 51];     // S padded row width 51 (col 50 = 0)
    __shared__ float inp_s[16 * 53];    // inp padded width 53 (cols 50..52 = 0)
    __shared__ float redF[4][16];
    __shared__ float redA[4][16];
    __shared__ float xs[16], beta_s[16], scT_s[16];
    __shared__ float aS[4][16], dxS[4][16];

    const int tid  = threadIdx.x;
    const int wv   = tid >> 5;
    const int lane = tid & 31;
    const int ln   = lane & 15;
    const int hb   = (lane < 16) ? 0 : 8;
    const int b0g  = blockIdx.x * 16;

    // stage S into LDS (coalesced), pad col 50 with zero
    for (int idx = tid; idx < 256 * INDIM; idx += 128) {
        int m = idx / INDIM, k = idx - m * INDIM;
        S_s[m * 51 + k] = S[idx];
    }
    for (int m = tid; m < 256; m += 128) S_s[m * 51 + 50] = 0.f;

    // gather Uvec into inp cols 2..49, zero pad cols 50..52
    for (int idx = tid; idx < 16 * 48; idx += 128) {
        int bl = idx / 48, j = idx - bl * 48;
        int u = j >> 4, r = j & 15;
        int iv = b_i_n[(b0g + bl) * 3 + u];
        const float* Uu = (u == 0) ? U0 : (u == 1) ? U1 : U2;
        inp_s[bl * 53 + 2 + j] = Uu[iv * 16 + r];
    }
    for (int idx = tid; idx < 16 * 3; idx += 128) {
        int bl = idx / 3, c = idx - bl * 3;
        inp_s[bl * 53 + 50 + c] = 0.f;
    }
    if (tid < 16) scT_s[tid] = b_t_n[b0g + tid];
    __syncthreads();

    if (tid < 16) {
        float acc = 0.f;
        for (int r = 0; r < 16; ++r)
            acc += inp_s[tid*53 + 2 + r] * inp_s[tid*53 + 18 + r] * inp_s[tid*53 + 34 + r];
        beta_s[tid] = acc; xs[tid] = acc;
    }
    __syncthreads();

    // per-wave constants: this wave owns m in [wv*64, wv*64+64)
    int   mro[4]; float w0r[4], w1r[4], s1r[4];
    #pragma unroll
    for (int nt = 0; nt < 4; ++nt) {
        int m = wv * 64 + nt * 16 + ln;
        mro[nt] = m; w0r[nt] = w[m]; w1r[nt] = w[m + 256]; s1r[nt] = S_s[m * 51 + 1];
    }
    float stv[8]; int bgv[8];
    #pragma unroll
    for (int v = 0; v < 8; ++v) { stv[v] = scT_s[hb + v]; bgv[v] = b0g + hb + v; }

    const int klo = (lane < 16) ? 0 : 2;

    for (int s = 0; s < 4; ++s) {
        if (tid < 16) {
            float ts = (s == 0) ? 0.f : ((s == 3) ? 1.f : 0.5f);
            inp_s[tid * 53 + 0] = scT_s[tid] * ts;
            inp_s[tid * 53 + 1] = xs[tid];
        }
        __syncthreads();

        // persist inp for this stage (width 64, zeros past 50)
        for (int idx = tid; idx < 16 * 64; idx += 128) {
            int bl = idx >> 6, k = idx & 63;
            ws[O_INP + (unsigned)((s * 512 + b0g + bl) * 64 + k)] = (k < 50) ? inp_s[bl * 53 + k] : 0.f;
        }

        // phi = inp (16x50) @ S^T (50x256) : WMMA f32 16x16x4, 13 K-steps x 4 N-tiles
        v8f acc[4];
        #pragma unroll
        for (int nt = 0; nt < 4; ++nt)
            #pragma unroll
            for (int v = 0; v < 8; ++v) acc[nt][v] = 0.f;

        #pragma unroll
        for (int kq = 0; kq < 13; ++kq) {
            int k0 = kq * 4 + klo;
            v2f af;
            af.x = inp_s[ln * 53 + k0];
            af.y = inp_s[ln * 53 + k0 + 1];
            #pragma unroll
            for (int nt = 0; nt < 4; ++nt) {
                v2f bf;
                bf.x = (k0     < 50) ? S_s[mro[nt] * 51 + k0]     : 0.f;
                bf.y = (k0 + 1 < 50) ? S_s[mro[nt] * 51 + k0 + 1] : 0.f;
                acc[nt] = __builtin_amdgcn_wmma_f32_16x16x4_f32(
                    false, af, false, bf, (short)0, acc[nt], false, false);
            }
        }

        // elementwise: sincos, F/a partials, persist d and phi
        float Fp[8], Ap[8];
        #pragma unroll
        for (int v = 0; v < 8; ++v) { Fp[v] = 0.f; Ap[v] = 0.f; }
        #pragma unroll
        for (int nt = 0; nt < 4; ++nt) {
            #pragma unroll
            for (int v = 0; v < 8; ++v) {
                float phi = acc[nt][v];
                float sn, cs;
                sincosf(phi, &sn, &cs);
                float fval  = w0r[nt] * cs + w1r[nt] * sn;
                float dterm = w1r[nt] * cs - w0r[nt] * sn;
                Fp[v] += fval;
                Ap[v] += dterm * s1r[nt];
                unsigned o = (unsigned)((s * 512 + bgv[v]) * 256 + mro[nt]);
                ws[O_D + o]   = stv[v] * dterm;
                ws[O_PHI + o] = phi;
            }
        }
        // deterministic reduction within each 16-lane half
        #pragma unroll
        for (int v = 0; v < 8; ++v) {
            float f = Fp[v], a = Ap[v];
            #pragma unroll
            for (int off = 8; off >= 1; off >>= 1) {
                f += __shfl_xor(f, off, 32);
                a += __shfl_xor(a, off, 32);
            }
            Fp[v] = f; Ap[v] = a;
        }
        if (ln == 0) {
            #pragma unroll
            for (int v = 0; v < 8; ++v) { redF[wv][hb + v] = Fp[v]; redA[wv][hb + v] = Ap[v]; }
        }
        __syncthreads();
        if (tid < 16) {
            float F  = redF[0][tid] + redF[1][tid] + redF[2][tid] + redF[3][tid];
            float A  = redA[0][tid] + redA[1][tid] + redA[2][tid] + redA[3][tid];
            float dx = scT_s[tid] * F;
            float a  = scT_s[tid] * A;
            dxS[s][tid] = dx; aS[s][tid] = a;
            if      (s == 0) xs[tid] = beta_s[tid] + 0.5f * dx;
            else if (s == 1) xs[tid] = beta_s[tid] + 0.5f * dx;
            else if (s == 2) xs[tid] = beta_s[tid] + dx;
        }
        __syncthreads();
    }

    if (tid < 16) {
        int bg = b0g + tid;
        float a1 = aS[0][tid], a2 = aS[1][tid], a3 = aS[2][tid], a4 = aS[3][tid];
        float dx1 = dxS[0][tid], dx2 = dxS[1][tid], dx3 = dxS[2][tid], dx4 = dxS[3][tid];
        float xf = beta_s[tid] + (dx1 + 2.f*dx2 + 2.f*dx3 + dx4) * (1.f/6.f);
        float gx = xf - b_obs[bg];
        float c1 = (1.f + a2 + 0.5f*a3*a2 + 0.25f*a4*a3*a2) * (1.f/6.f);
        float c2 = (2.f + a3 + 0.5f*a4*a3) * (1.f/6.f);
        float c3 = (2.f + a4) * (1.f/6.f);
        float c4 = (1.f/6.f);
        float kb1 = a1;
        float kb2 = a2 * (1.f + 0.5f*kb1);
        float kb3 = a3 * (1.f + 0.5f*kb2);
        float kb4 = a4 * (1.f + kb3);
        float gb = 1.f + (kb1 + 2.f*kb2 + 2.f*kb3 + kb4) * (1.f/6.f);
        const float inv = 1.f / 512.f;
        out[bg] = gx * gb * inv;                // beta-grad block
        ws[O_GX + bg] = gx; ws[O_GB + bg] = gb;
        float sc = scT_s[tid];
        float cc[4] = {c1, c2, c3, c4};
        #pragma unroll
        for (int s2 = 0; s2 < 4; ++s2) {
            float wvv = cc[s2] * gx * inv;
            ws[O_WV  + s2 * 512 + bg] = wvv;
            ws[O_WV2 + s2 * 512 + bg] = wvv * sc;
        }
    }
}

// S-grad: out_S(256x50) = sum_kt wvec[kt]*d[kt,m]*inp[kt,k], kt = 4*512. WMMA GEMM.
__global__ __launch_bounds__(32) void k2_sgrad(const float* __restrict__ ws, float* __restrict__ out)
{
    __shared__ float wv_s[2048];
    const int lane = threadIdx.x;
    const int mt = blockIdx.x >> 2;
    const int nt = blockIdx.x & 3;
    for (int i = lane; i < 2048; i += 32) wv_s[i] = ws[O_WV + i];
    __syncthreads();
    const int ln  = lane & 15;
    const int klo = (lane < 16) ? 0 : 2;
    const int m   = mt * 16 + ln;
    const int kc  = nt * 16 + ln;
    v8f acc;
    #pragma unroll
    for (int v = 0; v < 8; ++v) acc[v] = 0.f;
    for (int kq = 0; kq < 512; ++kq) {
        int kt = kq * 4 + klo;
        v2f af, bf;
        af.x = wv_s[kt]     * ws[O_D + (unsigned)kt * 256u + m];
        af.y = wv_s[kt + 1] * ws[O_D + (unsigned)(kt + 1) * 256u + m];
        bf.x = ws[O_INP + (unsigned)kt * 64u + kc];
        bf.y = ws[O_INP + (unsigned)(kt + 1) * 64u + kc];
        acc = __builtin_amdgcn_wmma_f32_16x16x4_f32(false, af, false, bf, (short)0, acc, false, false);
    }
    #pragma unroll
    for (int v = 0; v < 8; ++v) {
        int mm = mt * 16 + v + ((lane < 16) ? 0 : 8);
        int kk = nt * 16 + ln;
        if (kk < 50) out[25088 + mm * 50 + kk] = acc[v];
    }
}

// w-grad: out_w[m] = sum_kt wvec2[kt]*cos(phi[kt,m]);  out_w[256+m] with sin
__global__ __launch_bounds__(256) void k3_wgrad(const float* __restrict__ ws, float* __restrict__ out)
{
    const int tid = threadIdx.x;
    float ac = 0.f, as2 = 0.f;
    for (int kt = 0; kt < 2048; ++kt) {
        float w2  = ws[O_WV2 + kt];
        float phi = ws[O_PHI + (unsigned)kt * 256u + tid];
        float sn, cs; sincosf(phi, &sn, &cs);
        ac  = fmaf(w2, cs, ac);
        as2 = fmaf(w2, sn, as2);
    }
    out[37888 + tid]       = ac;
    out[37888 + 256 + tid] = as2;
}

// Uvec-grad: out[512 + b*48 + j] = gx/B*gb*prod_excl + sum_s wvec[s,b]*sum_m d[s,b,m]*S[m,2+j]
__global__ __launch_bounds__(64) void k4_ugrad(const float* __restrict__ ws,
                                               const float* __restrict__ S,
                                               float* __restrict__ out)
{
    __shared__ float dw[1024];
    __shared__ float uv[48];
    const int b = blockIdx.x;
    const int tid = threadIdx.x;
    for (int idx = tid; idx < 1024; idx += 64) {
        int s = idx >> 8, m = idx & 255;
        dw[idx] = ws[O_WV + s * 512 + b] * ws[O_D + (unsigned)((s * 512 + b) * 256 + m)];
    }
    if (tid < 48) uv[tid] = ws[O_INP + (unsigned)(b * 64 + 2 + tid)];  // stage 0 inp holds Uvec
    __syncthreads();
    if (tid < 48) {
        float acc = 0.f;
        for (int m = 0; m < 256; ++m) {
            float dsum = dw[m] + dw[256 + m] + dw[512 + m] + dw[768 + m];
            acc = fmaf(dsum, S[m * 50 + 2 + tid], acc);
        }
        int v = tid >> 4, r = tid & 15;
        float u0 = uv[r], u1 = uv[16 + r], u2 = uv[32 + r];
        float p = (v == 0) ? u1 * u2 : (v == 1) ? u0 * u2 : u0 * u1;
        float gx = ws[O_GX + b], gb = ws[O_GB + b];
        out[512 + b * 48 + tid] = gx * (1.f / 512.f) * gb * p + acc;
    }
}

extern "C" void kernel_launch(void* const* d_in, const int* in_sizes, int n_in,
                              void* d_out, int out_size, void* d_ws, size_t ws_size,
                              hipStream_t stream)
{
    (void)in_sizes; (void)n_in; (void)out_size; (void)ws_size;
    const float* U0   = (const float*)d_in[0];
    const float* U1   = (const float*)d_in[1];
    const float* U2   = (const float*)d_in[2];
    const float* S    = (const float*)d_in[3];
    const float* w    = (const float*)d_in[4];
    const float* btn  = (const float*)d_in[5];
    const float* bobs = (const float*)d_in[6];
    const int*   bin  = (const int*)d_in[7];
    float* out = (float*)d_out;
    float* ws  = (float*)d_ws;

    k1_stages<<<32, 128, 0, stream>>>(U0, U1, U2, S, w, btn, bobs, bin, ws, out);
    k2_sgrad<<<64, 32, 0, stream>>>(ws, out);
    k3_wgrad<<<1, 256, 0, stream>>>(ws, out);
    k4_ugrad<<<512, 64, 0, stream>>>(ws, S, out);
}